// Cross_Attention_19507741458643
// MI455X (gfx1250) — compile-verified
//
#include <hip/hip_runtime.h>
#include <hip/hip_bf16.h>

#define NNODES 50000
#define NEDGES 1600000
#define HD 64           // HEADS*DHEAD
#define NODE_TILES (NNODES / 16)   // 3125
#define EDGE_TILES (NEDGES / 16)   // 100000
#define RSQRT8 0.35355339059327373f

typedef __attribute__((ext_vector_type(2))) float v2f;
typedef __attribute__((ext_vector_type(8))) float v8f;

// D = A(16x4,f32) * B(4x16,f32) + C(16x16,f32)  -- CDNA5 fp32 WMMA
__device__ __forceinline__ v8f wmma4(v2f a, v2f b, v8f c) {
    return __builtin_amdgcn_wmma_f32_16x16x4_f32(
        /*neg_a=*/false, a, /*neg_b=*/false, b,
        /*c_mod=*/(short)0, c, /*reuse_a=*/false, /*reuse_b=*/false);
}

// ---------------- workspace layout (floats) ----------------
// Q:   [0,            3,200,000)
// K:   [3,200,000,    6,400,000)
// V:   [6,400,000,    9,600,000)
// s/e: [9,600,000,   22,400,000)   per (edge, head), biased score then exp
// m:   [22,400,000,  22,800,000)   biased segment max (>=0)
// den: [22,800,000,  23,200,000)
#define WS_Q   0
#define WS_K   3200000
#define WS_V   6400000
#define WS_S   9600000
#define WS_M   22400000
#define WS_DEN 22800000
// d_out: wV [0, 3,200,000)  then wE [3,200,000, 105,600,000)
#define OUT_WE 3200000

__global__ void init_kernel(float* __restrict__ wV, float* __restrict__ mB,
                            float* __restrict__ den) {
    size_t t = (size_t)blockIdx.x * blockDim.x + threadIdx.x;
    if (t < (size_t)NNODES * HD) wV[t] = 0.0f;
    if (t < (size_t)NNODES * 8) { mB[t] = 0.0f; den[t] = 0.0f; }
}

// ---- Q/K/V node GEMM: one wave handles 16 node rows, 16x64 @ 64x64 ----
__global__ void qkv_kernel(const float* __restrict__ x,
                           const float* __restrict__ WQ, const float* __restrict__ bQ,
                           const float* __restrict__ WK, const float* __restrict__ WV,
                           float* __restrict__ Q, float* __restrict__ K,
                           float* __restrict__ V) {
    const int lane = threadIdx.x & 31;
    const int half = lane >> 4;      // K sub-block selector for A/B layout
    const int ln   = lane & 15;      // M index (A) / N index (B,C,D)
    int tile = blockIdx.x * (blockDim.x >> 5) + (threadIdx.x >> 5);
    if (tile >= NODE_TILES) return;  // wave-uniform: EXEC stays all-1 for WMMA
    const size_t rowBase = (size_t)tile * 16;

#pragma unroll
    for (int m = 0; m < 3; ++m) {
        const float* __restrict__ W = (m == 0) ? WQ : (m == 1) ? WK : WV;
        float* __restrict__ O       = (m == 0) ? Q  : (m == 1) ? K  : V;
        v8f acc[4] = {};
        for (int k = 0; k < 64; k += 4) {
            // A: lane ln holds row rowBase+ln, cols k+2*half .. +1
            v2f a = *(const v2f*)(x + (rowBase + ln) * 64 + k + 2 * half);
            const int kk = k + 2 * half;
#pragma unroll
            for (int nt = 0; nt < 4; ++nt) {
                v2f b;
                b.x = W[(size_t)kk * 64 + nt * 16 + ln];
                b.y = W[(size_t)(kk + 1) * 64 + nt * 16 + ln];
                acc[nt] = wmma4(a, b, acc[nt]);
            }
        }
#pragma unroll
        for (int nt = 0; nt < 4; ++nt) {
            const int col = nt * 16 + ln;
            const float bias = (m == 0) ? bQ[col] : 0.0f;
#pragma unroll
            for (int r = 0; r < 8; ++r) {
                size_t row = rowBase + r + 8 * half;  // C layout: vgpr r, halves
                O[row * 64 + col] = acc[nt][r] + bias;
            }
        }
    }
}

// ---- Edge kernel: E GEMM (WMMA) fused with gather + score + seg-max ----
__global__ void edge_score_kernel(const float* __restrict__ ea,
                                  const float* __restrict__ WE1,
                                  const float* __restrict__ bE1,
                                  const int* __restrict__ eidx,  // [2, NEDGES]
                                  const float* __restrict__ Qp,
                                  const float* __restrict__ Kp,
                                  float* __restrict__ sW,   // biased scores [NEDGES*8]
                                  float* __restrict__ mB,   // biased max [NNODES*8]
                                  float* __restrict__ wE) { // d_out + OUT_WE
    const int lane = threadIdx.x & 31;
    const int half = lane >> 4;
    const int ln   = lane & 15;
    const int tile = blockIdx.x * (blockDim.x >> 5) + (threadIdx.x >> 5);
    const size_t rowBase = (size_t)tile * 16;
    const int* __restrict__ srcI = eidx;
    const int* __restrict__ dstI = eidx + NEDGES;

    v8f acc[4] = {};
    for (int k = 0; k < 64; k += 4) {
        // streaming read: non-temporal so Q/K/V stay L2-resident
        v2f a = __builtin_nontemporal_load(
            (const v2f*)(ea + (rowBase + ln) * 64 + k + 2 * half));
        const int kk = k + 2 * half;
#pragma unroll
        for (int nt = 0; nt < 4; ++nt) {
            v2f b;
            b.x = WE1[(size_t)kk * 64 + nt * 16 + ln];
            b.y = WE1[(size_t)(kk + 1) * 64 + nt * 16 + ln];
            acc[nt] = wmma4(a, b, acc[nt]);
        }
    }

    int se[8], de[8];
#pragma unroll
    for (int r = 0; r < 8; ++r) {
        size_t row = rowBase + r + 8 * half;
        se[r] = srcI[row];
        de[r] = dstI[row];
    }

#pragma unroll
    for (int nt = 0; nt < 4; ++nt) {
        const int col  = nt * 16 + ln;
        const float bias = bE1[col];
        const int head = nt * 2 + (ln >> 3);
#pragma unroll
        for (int r = 0; r < 8; ++r) {
            const size_t row = rowBase + r + 8 * half;
            const float Ev = acc[nt][r] + bias;
            const float kv = Kp[(size_t)se[r] * 64 + col];   // L2-resident gather
            const float qv = Qp[(size_t)de[r] * 64 + col];   // L2-resident gather
            const float sc = kv * qv * Ev;          // score_e == wE
            __builtin_nontemporal_store(sc, &wE[row * 64 + col]);  // 410MB stream
            // reduce over the 8 lanes covering one head (d-dimension)
            float sum = sc;
            sum += __shfl_xor(sum, 1);
            sum += __shfl_xor(sum, 2);
            sum += __shfl_xor(sum, 4);
            if ((lane & 7) == 0) {
                float sv = fminf(fmaxf(sum * RSQRT8, -5.0f), 5.0f) + 5.0f; // bias>=0
                sW[row * 8 + head] = sv;
                // int atomicMax == float max for non-negative floats
                atomicMax((int*)(mB + (size_t)de[r] * 8 + head),
                          __float_as_int(sv));
            }
        }
    }
}

// ---- softmax pass 2: e = exp(s - m), accumulate denominator ----
__global__ void exp_den_kernel(const int* __restrict__ dstI,
                               float* __restrict__ sW,
                               const float* __restrict__ mB,
                               float* __restrict__ den) {
    size_t t = (size_t)blockIdx.x * blockDim.x + threadIdx.x;  // < NEDGES*8
    size_t e = t >> 3;
    int h = (int)(t & 7);
    int d = dstI[e];
    float ev = __expf(sW[t] - mB[(size_t)d * 8 + h]);  // bias cancels
    sW[t] = ev;
    atomicAdd(&den[(size_t)d * 8 + h], ev);
}

// ---- aggregation: wV[dst] += (V[src] + score_e) * alpha ----
__global__ void aggregate_kernel(const int* __restrict__ eidx,
                                 const float* __restrict__ Vp,
                                 const float* __restrict__ eW,  // exp values
                                 const float* __restrict__ den,
                                 const float* __restrict__ wE,
                                 float* __restrict__ wV) {
    size_t t = (size_t)blockIdx.x * blockDim.x + threadIdx.x;  // < NEDGES*64
    size_t e = t >> 6;
    int c = (int)(t & 63);
    int h = c >> 3;
    int s = eidx[e];
    int d = eidx[NEDGES + e];
    float alpha = eW[e * 8 + h] / (den[(size_t)d * 8 + h] + 1e-16f);
    float we = __builtin_nontemporal_load(&wE[t]);   // 410MB stream re-read
    float val = (Vp[(size_t)s * 64 + c] + we) * alpha;
    atomicAdd(&wV[(size_t)d * 64 + c], val);
}

extern "C" void kernel_launch(void* const* d_in, const int* in_sizes, int n_in,
                              void* d_out, int out_size, void* d_ws, size_t ws_size,
                              hipStream_t stream) {
    const float* x    = (const float*)d_in[0];
    const float* ea   = (const float*)d_in[1];
    const float* WQ   = (const float*)d_in[2];
    const float* bQ   = (const float*)d_in[3];
    const float* WK   = (const float*)d_in[4];
    const float* WV   = (const float*)d_in[5];
    const float* WE1  = (const float*)d_in[6];
    const float* bE1  = (const float*)d_in[7];
    const int*   eidx = (const int*)d_in[8];

    float* out = (float*)d_out;
    float* ws  = (float*)d_ws;
    float* Q   = ws + WS_Q;
    float* K   = ws + WS_K;
    float* V   = ws + WS_V;
    float* sW  = ws + WS_S;
    float* mB  = ws + WS_M;
    float* den = ws + WS_DEN;
    float* wV  = out;
    float* wE  = out + OUT_WE;

    // 1) zero wV / m / den
    init_kernel<<<((size_t)NNODES * HD + 255) / 256, 256, 0, stream>>>(wV, mB, den);

    // 2) node Q/K/V GEMMs (WMMA f32 16x16x4)
    {
        int waves = 4;  // blockDim 128 = 4 waves, one 16-row tile each
        int blocks = (NODE_TILES + waves - 1) / waves;
        qkv_kernel<<<blocks, 32 * waves, 0, stream>>>(x, WQ, bQ, WK, WV, Q, K, V);
    }

    // 3) edge GEMM + score + wE + segment max (EDGE_TILES divisible by 4)
    edge_score_kernel<<<EDGE_TILES / 4, 128, 0, stream>>>(
        ea, WE1, bE1, eidx, Q, K, sW, mB, wE);

    // 4) exp + denominator
    exp_den_kernel<<<((size_t)NEDGES * 8) / 256, 256, 0, stream>>>(
        eidx + NEDGES, sW, mB, den);

    // 5) aggregation into wV
    aggregate_kernel<<<((size_t)NEDGES * 64) / 256, 256, 0, stream>>>(
        eidx, V, sW, den, wE, wV);
}